// CTCLayer_89026082111753
// MI455X (gfx1250) — compile-verified
//
#include <hip/hip_runtime.h>
#include <stdint.h>

// CTC loss (tf.nn.ctc_loss semantics, blank_index=0) for B=1024, T=256, C=128, L=32.
// One wave32 per batch element; alpha DP entirely in registers + shuffles;
// y_pred rows streamed global->LDS with CDNA5 async-to-LDS double buffering.

namespace {
constexpr int Tt = 256;
constexpr int Cc = 128;
constexpr int Ll = 32;
constexpr int PADv = 127;
constexpr float NEGV = -1e30f;

__device__ __forceinline__ float lae(float a, float b) {
  // logaddexp; exp underflows to 0 for (n-m) <= -104, so NEGV is exact log-zero
  float m = fmaxf(a, b);
  float n = fminf(a, b);
  return m + log1pf(__expf(n - m));
}

__device__ __forceinline__ float wmax(float v) {
#pragma unroll
  for (int off = 16; off > 0; off >>= 1) v = fmaxf(v, __shfl_xor(v, off, 32));
  return v;
}
__device__ __forceinline__ float wsum(float v) {
#pragma unroll
  for (int off = 16; off > 0; off >>= 1) v += __shfl_xor(v, off, 32);
  return v;
}
}  // namespace

__global__ __launch_bounds__(32) void ctc_alpha_kernel(
    const int* __restrict__ y_true, const float* __restrict__ y_pred,
    float* __restrict__ out) {
  __shared__ __align__(16) float buf[2][Cc];   // double-buffered logit row
  __shared__ float afin[2 * Ll + 1];           // final alpha for readout

  const int b = blockIdx.x;
  const int lane = threadIdx.x;  // 0..31, one wave per block

  // Labels: lane l owns label l (ext state s=2l+1) and states s=2l, 2l+1.
  const int raw = y_true[b * Ll + lane];
  const int lab = raw & (Cc - 1);
  const int labp = __shfl_up(lab, 1, 32);
  const bool skip_ok = (lane > 0) && (lab != labp);  // s-2 -> s legal for odd s

  const float* base = y_pred + (size_t)b * Tt * Cc + 4 * lane;
  // Generic LDS pointer: low 32 bits are the wave-relative LDS byte address.
  const uint32_t lds0 = (uint32_t)(uintptr_t)&buf[0][4 * lane];
  const uint32_t lds1 = (uint32_t)(uintptr_t)&buf[1][4 * lane];

  // ---- CDNA5 async global->LDS (ASYNCcnt-tracked): VDST = LDS byte addr,
  //      VADDR = 64-bit global addr. 32 lanes x b128 = one 512B logit row. ----
#define ASYNC_LOAD_ROW(t, ldsoff)                                          \
  asm volatile("global_load_async_to_lds_b128 %0, %1, off"                 \
               :: "v"(ldsoff),                                             \
                  "v"((uint64_t)(uintptr_t)(base + (size_t)(t) * Cc))      \
               : "memory")
#define WAIT_ASYNC_LE1() asm volatile("s_wait_asynccnt 0x1" ::: "memory")

  ASYNC_LOAD_ROW(0, lds0);
  ASYNC_LOAD_ROW(1, lds1);
  WAIT_ASYNC_LE1();  // async loads complete in order -> row 0 resident

  // ---- t = 0 init: alpha0[0]=lp(blank), alpha0[1]=lp(lab[0]), rest NEG ----
  float aE, aO, a64 = NEGV;
  {
    const float4 p = *reinterpret_cast<const float4*>(&buf[0][4 * lane]);
    float mx = wmax(fmaxf(fmaxf(p.x, p.y), fmaxf(p.z, p.w)));
    float s = wsum(__expf(p.x - mx) + __expf(p.y - mx) + __expf(p.z - mx) +
                   __expf(p.w - mx));
    const float lse = mx + __logf(s);
    const float p0 = buf[0][0];
    const float pl = buf[0][lab];
    aE = (lane == 0) ? (p0 - lse) : NEGV;
    aO = (lane == 0) ? (pl - lse) : NEGV;
  }

  // ---- alpha recurrence over t = 1..T-1, prefetching row t+1 ----
  for (int t = 1; t < Tt; ++t) {
    const int tn = (t + 1 < Tt) ? (t + 1) : (Tt - 1);  // harmless dup at tail
    ASYNC_LOAD_ROW(tn, ((t + 1) & 1) ? lds1 : lds0);
    WAIT_ASYNC_LE1();  // row t resident

    const int w = t & 1;
    const float4 p = *reinterpret_cast<const float4*>(&buf[w][4 * lane]);
    float mx = wmax(fmaxf(fmaxf(p.x, p.y), fmaxf(p.z, p.w)));
    float s = wsum(__expf(p.x - mx) + __expf(p.y - mx) + __expf(p.z - mx) +
                   __expf(p.w - mx));
    const float lse = mx + __logf(s);
    const float lpb = buf[w][0] - lse;    // blank log-prob (uniform, LDS bcast)
    const float lpl = buf[w][lab] - lse;  // this lane's label log-prob

    float prevO = __shfl_up(aO, 1, 32);   // alpha[2l-1]
    if (lane == 0) prevO = NEGV;
    const float a63 = __shfl(aO, 31, 32); // alpha[63] for state 64

    // even s=2l: lp_blank + lae(alpha[s], alpha[s-1]); no skip into blanks
    const float nE = lpb + lae(aE, prevO);
    // odd s=2l+1: lp_lab + lae(lae(alpha[s], alpha[s-1]), skip? alpha[s-2])
    const float nO = lpl + lae(lae(aO, aE), skip_ok ? prevO : NEGV);
    // s=64 (trailing blank), replicated on all lanes
    const float n64 = lpb + lae(a64, a63);
    aE = nE; aO = nO; a64 = n64;
  }

  // ---- readout: loglik = lae(alpha[2*len], alpha[2*len-1]) ----
  afin[2 * lane] = aE;
  afin[2 * lane + 1] = aO;
  if (lane == 0) afin[2 * Ll] = a64;
  __builtin_amdgcn_wave_barrier();  // single wave: DS ops are in-order anyway
  if (lane == 0) {
    const int len = (int)__popcll(__ballot(raw != PADv));
    const int e1 = 2 * len;
    const int e2 = (2 * len - 1 < 0) ? 0 : (2 * len - 1);
    out[b] = -lae(afin[e1], afin[e2]);
  }
#undef ASYNC_LOAD_ROW
#undef WAIT_ASYNC_LE1
}

extern "C" void kernel_launch(void* const* d_in, const int* in_sizes, int n_in,
                              void* d_out, int out_size, void* d_ws, size_t ws_size,
                              hipStream_t stream) {
  (void)n_in; (void)out_size; (void)d_ws; (void)ws_size;
  const int* y_true = (const int*)d_in[0];
  const float* y_pred = (const float*)d_in[1];
  float* out = (float*)d_out;
  const int Bn = in_sizes[0] / Ll;  // 1024
  ctc_alpha_kernel<<<dim3(Bn), dim3(32), 0, stream>>>(y_true, y_pred, out);
}